// Decoder_5669356831490
// MI455X (gfx1250) — compile-verified
//
#include <hip/hip_runtime.h>
#include <hip/hip_bf16.h>

// est_source: [B=16, C=2, D2=256, F=4000] f32  (frames contiguous)
// out:        [B=16, C=2, 32008] f32, out[bc, 8s+r] = P[s,r] + P[s-1,8+r]
// P[f,k] = (1/16) * sum_{l<16} est[bc, 16k+l, f]  computed via v_wmma_f32_16x16x4_f32

typedef __attribute__((ext_vector_type(2))) float v2f;
typedef __attribute__((ext_vector_type(8))) float v8f;

#define FRAMES       4000
#define D2V          256
#define OUT_T        4001          // output subframes per (b,c): 8*(F-1)+16 = 32008 = 8*4001
#define OUT_LEN      32008
#define TILE_OUT     15            // outputs per wave (16-frame P tile with 1-frame halo)
#define TILES_PER_BC 267           // ceil(4001/15)
#define N_BC         32            // 16*2
#define WAVES_PER_BLOCK 8
#define N_BLOCKS     ((N_BC * TILES_PER_BC) / WAVES_PER_BLOCK)   // 8544/8 = 1068

__global__ __launch_bounds__(256)
void decoder_oadd_wmma(const float* __restrict__ est, float* __restrict__ out)
{
    // per-wave P tile in LDS: [frameOff 0..15][window 0..15], padded stride 17
    __shared__ float lds[WAVES_PER_BLOCK][16][17];

    const int lane  = threadIdx.x & 31;
    const int wave  = threadIdx.x >> 5;
    const int gwave = blockIdx.x * WAVES_PER_BLOCK + wave;   // exactly 32*267 waves
    const int bc    = gwave / TILES_PER_BC;                  // 0..31
    const int tile  = gwave % TILES_PER_BC;

    const int f0 = tile * TILE_OUT - 1;      // first P frame of tile (may be -1)
    const int m  = lane & 15;                // WMMA A row (frame offset in tile); B/C col (window)
    const int kg = lane >> 4;                // 0/1 -> A K-base = 2*kg
    const int frame = f0 + m;                // per-lane global frame (constant over loop)

    const float* srcbc = est + (size_t)bc * (size_t)(D2V * FRAMES);

    v8f acc = {};
    if (f0 >= 0 && f0 + 15 < FRAMES) {
        // ---- hot path (wave-uniform: all 16 frames valid) -------------------
        // lane-fixed column pointer: d2 = 2*kg, frame column = frame
        const float* p0 = srcbc + (size_t)(2 * kg) * FRAMES + frame;
        #pragma unroll 4
        for (int c4 = 0; c4 < 64; ++c4) {
            v2f a;
            a.x = p0[(size_t)(4 * c4)     * FRAMES];   // d2 = 4c4 + 2kg
            a.y = p0[(size_t)(4 * c4 + 1) * FRAMES];   // d2 = 4c4 + 2kg + 1
            const float bv = (m == (c4 >> 2)) ? 0.0625f : 0.0f;  // hoisted per unroll group
            v2f b; b.x = bv; b.y = bv;
            acc = __builtin_amdgcn_wmma_f32_16x16x4_f32(
                false, a, false, b, (short)0, acc, false, false);
        }
    } else {
        // ---- boundary path (tile 0 and the tail tiles) ----------------------
        const bool fvalid = (frame >= 0) && (frame < FRAMES);
        #pragma unroll 4
        for (int c4 = 0; c4 < 64; ++c4) {
            v2f a; a.x = 0.0f; a.y = 0.0f;
            if (fvalid) {
                const float* p = srcbc + (size_t)(4 * c4 + 2 * kg) * FRAMES + frame;
                a.x = p[0];
                a.y = p[FRAMES];
            }
            const float bv = (m == (c4 >> 2)) ? 0.0625f : 0.0f;
            v2f b; b.x = bv; b.y = bv;
            acc = __builtin_amdgcn_wmma_f32_16x16x4_f32(
                false, a, false, b, (short)0, acc, false, false);
        }
    }

    // C layout: VGPR v -> row M = v + 8*kg (frame offset), col N = m (window)
    #pragma unroll
    for (int v = 0; v < 8; ++v)
        lds[wave][v + 8 * kg][m] = acc[v];

    __syncthreads();

    // Epilogue: outputs s = f0+1 .. f0+15  (frameOff 1..15 in the P tile)
    //   out[bc, 8s+r] = P[off, r] + P[off-1, 8+r]
    // 120 contiguous floats per wave starting at 8*(f0+1).
    float* obase = out + (size_t)bc * OUT_LEN;
    for (int i = lane; i < 120; i += 32) {
        const int off = 1 + (i >> 3);
        const int r   = i & 7;
        const int s   = f0 + off;            // >= 0 always; clip top edge (last tile)
        if (s < OUT_T)
            obase[8 * s + r] = lds[wave][off][r] + lds[wave][off - 1][8 + r];
    }
}

extern "C" void kernel_launch(void* const* d_in, const int* in_sizes, int n_in,
                              void* d_out, int out_size, void* d_ws, size_t ws_size,
                              hipStream_t stream)
{
    const float* est = (const float*)d_in[0];
    float* out = (float*)d_out;
    hipLaunchKernelGGL(decoder_oadd_wmma,
                       dim3(N_BLOCKS), dim3(256), 0, stream,
                       est, out);
}